// Pooling_64037962383970
// MI455X (gfx1250) — compile-verified
//
#include <hip/hip_runtime.h>
#include <cstdint>

// ---------------------------------------------------------------------------
// BN(eval) + ELU + segment-mean over sorted ids.
//   x: [N,128] f32, ids: [N] i32 sorted, out: [num_seg,128] f32
// One wave32 block per segment; rows streamed global->LDS with CDNA5 async
// copies (ASYNCcnt, th:NT since x is stream-once), double-buffered so the DMA
// of chunk k+1 overlaps BN+ELU+accumulate of chunk k. Sorted ids -> binary
// search bounds, zero atomics, each output written exactly once.
// ---------------------------------------------------------------------------

static constexpr int kF      = 128;          // features
static constexpr int kLanes  = 32;           // wave32
static constexpr int kChunk  = 8;            // rows staged per chunk
static constexpr int kRowB   = kF * 4;       // 512 B per row
static constexpr int kChunkB = kChunk * kRowB;

#define BN_EPS 1e-5f

// CDNA5 async copy (GVS): copies 16 B per lane from s_base + vaddr + offset into LDS at vdst + offset
// (same IOFFSET is added to BOTH the LDS and the global address, so one
//  address pair + immediate offsets covers a whole chunk of rows)
__device__ __forceinline__ void async_row_load_dyn(uint32_t lds_addr,
                                                   const float* sbase,
                                                   uint32_t voff) {
  asm volatile("global_load_async_to_lds_b128 %0, %1, %2 th:TH_LOAD_NT"
               :
               : "v"(lds_addr), "v"(voff), "s"(sbase));
}

template <int OFF>
__device__ __forceinline__ void async_row_load_imm(uint32_t lds_addr,
                                                   const float* sbase,
                                                   uint32_t voff) {
  asm volatile("global_load_async_to_lds_b128 %0, %1, %2 offset:%3 th:TH_LOAD_NT"
               :
               : "v"(lds_addr), "v"(voff), "s"(sbase), "n"(OFF));
}

__device__ __forceinline__ void issue_full_chunk(uint32_t lds_addr,
                                                 const float* x,
                                                 uint32_t voff) {
  async_row_load_imm<0 * kRowB>(lds_addr, x, voff);
  async_row_load_imm<1 * kRowB>(lds_addr, x, voff);
  async_row_load_imm<2 * kRowB>(lds_addr, x, voff);
  async_row_load_imm<3 * kRowB>(lds_addr, x, voff);
  async_row_load_imm<4 * kRowB>(lds_addr, x, voff);
  async_row_load_imm<5 * kRowB>(lds_addr, x, voff);
  async_row_load_imm<6 * kRowB>(lds_addr, x, voff);
  async_row_load_imm<7 * kRowB>(lds_addr, x, voff);
}

__device__ __forceinline__ void wait_async0() {
  asm volatile("s_wait_asynccnt 0" ::: "memory");
}

// ELU via hardware transcendental: v_mul + v_exp_f32 (TRANS co-issues) + add.
__device__ __forceinline__ float elu1(float t) {
  return t > 0.f ? t : (__expf(t) - 1.f);
}

__device__ __forceinline__ int lower_bound_i32(const int* __restrict__ a,
                                               int n, int key) {
  int lo = 0, hi = n;
  while (lo < hi) {
    int mid = (lo + hi) >> 1;
    if (a[mid] < key) lo = mid + 1; else hi = mid;
  }
  return lo;
}

__global__ __launch_bounds__(kLanes)
void seg_mean_bn_elu(const float* __restrict__ x,
                     const float* __restrict__ gamma,
                     const float* __restrict__ beta,
                     const float* __restrict__ rmean,
                     const float* __restrict__ rvar,
                     const int* __restrict__ ids,
                     float* __restrict__ out,
                     int n_rows) {
  __shared__ float4 smem[2 * kChunk * kLanes];  // 8 KB double buffer

  const int lane = threadIdx.x;   // lane owns features 4*lane .. 4*lane+3
  const int seg  = blockIdx.x;

  // Fold BN to x*scale + shift (per-lane float4 of params, coalesced loads).
  const float4 g = reinterpret_cast<const float4*>(gamma)[lane];
  const float4 b = reinterpret_cast<const float4*>(beta)[lane];
  const float4 m = reinterpret_cast<const float4*>(rmean)[lane];
  const float4 v = reinterpret_cast<const float4*>(rvar)[lane];
  float4 sc, sh;
  sc.x = g.x * rsqrtf(v.x + BN_EPS); sh.x = b.x - m.x * sc.x;
  sc.y = g.y * rsqrtf(v.y + BN_EPS); sh.y = b.y - m.y * sc.y;
  sc.z = g.z * rsqrtf(v.z + BN_EPS); sh.z = b.z - m.z * sc.z;
  sc.w = g.w * rsqrtf(v.w + BN_EPS); sh.w = b.w - m.w * sc.w;

  // Sorted ids -> contiguous row range for this segment (uniform -> SMEM).
  const int start = lower_bound_i32(ids, n_rows, seg);
  const int end   = lower_bound_i32(ids, n_rows, seg + 1);
  const int rows  = end - start;

  // Low 32 bits of a flat shared-aperture address are the LDS byte offset.
  const uint32_t lds_base = (uint32_t)(uintptr_t)(&smem[0]);
  const uint32_t lane_b   = (uint32_t)lane * 16u;

  float4 acc = make_float4(0.f, 0.f, 0.f, 0.f);

  const int nchunks = (rows + kChunk - 1) / kChunk;

  // Prefetch chunk 0.
  {
    const uint32_t dst = lds_base + lane_b;
    const uint32_t src = (uint32_t)start * (uint32_t)kRowB + lane_b;
    if (rows >= kChunk) {
      issue_full_chunk(dst, x, src);
    } else {
      for (int r = 0; r < rows; ++r)
        async_row_load_dyn(dst + (uint32_t)(r * kRowB), x,
                           src + (uint32_t)(r * kRowB));
    }
  }

  for (int c = 0; c < nchunks; ++c) {
    wait_async0();  // chunk c is now resident in LDS

    // Kick off chunk c+1 into the other buffer (overlaps compute below).
    const int nrem = rows - (c + 1) * kChunk;
    const uint32_t ndst = lds_base + (uint32_t)(((c + 1) & 1) * kChunkB) + lane_b;
    const uint32_t nsrc = (uint32_t)(start + (c + 1) * kChunk) * (uint32_t)kRowB + lane_b;
    if (nrem >= kChunk) {
      issue_full_chunk(ndst, x, nsrc);
    } else {
      for (int r = 0; r < nrem; ++r)
        async_row_load_dyn(ndst + (uint32_t)(r * kRowB), x,
                           nsrc + (uint32_t)(r * kRowB));
    }

    // BN + ELU + accumulate chunk c.
    const int cnt = (rows - c * kChunk) < kChunk ? (rows - c * kChunk) : kChunk;
    const float4* __restrict__ buf = &smem[(c & 1) * (kChunk * kLanes)];
    if (cnt == kChunk) {
#pragma unroll
      for (int r = 0; r < kChunk; ++r) {
        const float4 xv = buf[r * kLanes + lane];
        acc.x += elu1(xv.x * sc.x + sh.x);
        acc.y += elu1(xv.y * sc.y + sh.y);
        acc.z += elu1(xv.z * sc.z + sh.z);
        acc.w += elu1(xv.w * sc.w + sh.w);
      }
    } else {
      for (int r = 0; r < cnt; ++r) {
        const float4 xv = buf[r * kLanes + lane];
        acc.x += elu1(xv.x * sc.x + sh.x);
        acc.y += elu1(xv.y * sc.y + sh.y);
        acc.z += elu1(xv.z * sc.z + sh.z);
        acc.w += elu1(xv.w * sc.w + sh.w);
      }
    }
  }

  // scatter_mean semantics: empty segments -> 0 (count clamped to 1).
  const float inv = 1.0f / (float)(rows > 0 ? rows : 1);
  const float4 res = make_float4(acc.x * inv, acc.y * inv,
                                 acc.z * inv, acc.w * inv);
  reinterpret_cast<float4*>(out)[seg * kLanes + lane] = res;
}

extern "C" void kernel_launch(void* const* d_in, const int* in_sizes, int n_in,
                              void* d_out, int out_size, void* d_ws, size_t ws_size,
                              hipStream_t stream) {
  const float* x     = (const float*)d_in[0];
  const float* gamma = (const float*)d_in[1];
  const float* beta  = (const float*)d_in[2];
  const float* rmean = (const float*)d_in[3];
  const float* rvar  = (const float*)d_in[4];
  const int*   ids   = (const int*)d_in[5];
  float*       out   = (float*)d_out;

  const int n_rows  = in_sizes[0] / kF;   // 1,000,000
  const int num_seg = out_size   / kF;    // 50,000

  dim3 grid((unsigned)num_seg), block(kLanes);
  seg_mean_bn_elu<<<grid, block, 0, stream>>>(x, gamma, beta, rmean, rvar,
                                              ids, out, n_rows);
}